// label_SVDD_72043781423170
// MI455X (gfx1250) — compile-verified
//
#include <hip/hip_runtime.h>

typedef __attribute__((ext_vector_type(2))) float v2f;
typedef __attribute__((ext_vector_type(8))) float v8f;

#define N_NODES 100000
#define N_EDGES 3200000
#define IN_DIM  256
#define HID     16
#define EMBD    16
#define NCLS    2

// ---- workspace layout (in floats) ----
#define WS_OUTDEG 0
#define WS_INDEG  (N_NODES)
#define WS_ROUT   (2 * N_NODES)
#define WS_RIN    (3 * N_NODES)
#define WS_H0     (4 * N_NODES)
#define WS_AGG1   (WS_H0   + N_NODES * HID)
#define WS_H2     (WS_AGG1 + N_NODES * HID)
#define WS_AGG2   (WS_H2   + N_NODES * HID)
#define WS_TOTAL  (WS_AGG2 + N_NODES * HID)   // 6.8M floats = 27.2 MB

// ---------------- utility: zero scratch ----------------
__global__ void k_zero(float* __restrict__ p, int n) {
  int i = blockIdx.x * blockDim.x + threadIdx.x;
  if (i < n) p[i] = 0.0f;
}

// ---------------- degree accumulation ----------------
__global__ void k_degrees(const int* __restrict__ src, const int* __restrict__ dst,
                          float* __restrict__ outdeg, float* __restrict__ indeg) {
  int e = blockIdx.x * blockDim.x + threadIdx.x;
  if (e < N_EDGES) {
    atomicAdd(&outdeg[src[e]], 1.0f);
    atomicAdd(&indeg[dst[e]], 1.0f);
  }
}

// ---------------- rsqrt of clamped degrees ----------------
__global__ void k_rsqrt(const float* __restrict__ outdeg, const float* __restrict__ indeg,
                        float* __restrict__ rout, float* __restrict__ rin) {
  int i = blockIdx.x * blockDim.x + threadIdx.x;
  if (i < N_NODES) {
    rout[i] = rsqrtf(fmaxf(outdeg[i], 1.0f));
    rin[i]  = rsqrtf(fmaxf(indeg[i], 1.0f));
  }
}

// ---------------- GEMM1: H0 = (X * rout) @ W0  (100000x256 @ 256x16) ----------------
// One wave per 16-row tile, full-precision v_wmma_f32_16x16x4_f32, 64 K-steps.
__global__ void k_gemm1_wmma(const float* __restrict__ X, const float* __restrict__ W0,
                             const float* __restrict__ rout, float* __restrict__ H0) {
  const int wave = (blockIdx.x * blockDim.x + threadIdx.x) >> 5;
  const int lane = threadIdx.x & 31;
  const int row0 = wave * 16;
  if (row0 >= N_NODES) return;            // wave-uniform: EXEC stays all-ones for WMMA
  const int m    = lane & 15;
  const int half = lane >> 4;             // K-chunk: half h covers K = 2h, 2h+1
  const int row  = row0 + m;
  const float s  = rout[row];
  const float* xr = X + (size_t)row * IN_DIM;

  v8f acc = {};
  for (int k = 0; k < IN_DIM; k += 4) {
    // A (16x4): lane(m,h) holds X[row][k+2h], X[row][k+2h+1], pre-scaled by rsqrt(out_deg)
    const float2 xa = *(const float2*)(xr + k + 2 * half);
    v2f a; a[0] = xa.x * s; a[1] = xa.y * s;
    // B (4x16): lane(n=m,h) holds W0[k+2h][n], W0[k+2h+1][n]  (coalesced 64B per half-wave)
    v2f b; b[0] = W0[(k + 2 * half) * HID + m];
           b[1] = W0[(k + 2 * half + 1) * HID + m];
    acc = __builtin_amdgcn_wmma_f32_16x16x4_f32(false, a, false, b, (short)0, acc,
                                                false, false);
  }
  // C/D layout: VGPR i -> M = i + 8*half, N = m
  float* outp = H0 + (size_t)(row0 + 8 * half) * HID + m;
#pragma unroll
  for (int i = 0; i < 8; ++i) outp[(size_t)i * HID] = acc[i];
}

// ---------------- edge scatter: agg[dst] += H[src]  (16-wide rows, float4 chunks) ----------------
__global__ void k_scatter(const float* __restrict__ H, const int* __restrict__ src,
                          const int* __restrict__ dst, float* __restrict__ agg) {
  int t = blockIdx.x * blockDim.x + threadIdx.x;
  if (t >= N_EDGES * 4) return;
  const int e = t >> 2;
  const int q = (t & 3) * 4;
  const float4 v = *(const float4*)(H + (size_t)src[e] * HID + q);
  float* a = agg + (size_t)dst[e] * HID + q;
  atomicAdd(a + 0, v.x);
  atomicAdd(a + 1, v.y);
  atomicAdd(a + 2, v.z);
  atomicAdd(a + 3, v.w);
}

// ---------------- GEMM2 (fused layer-1 epilogue): H2 = ((agg1*rin + b0) * rout) @ W1 ----------------
__global__ void k_gemm2_wmma(const float* __restrict__ AGG1, const float* __restrict__ W1,
                             const float* __restrict__ b0, const float* __restrict__ rin,
                             const float* __restrict__ rout, float* __restrict__ H2) {
  const int wave = (blockIdx.x * blockDim.x + threadIdx.x) >> 5;
  const int lane = threadIdx.x & 31;
  const int row0 = wave * 16;
  if (row0 >= N_NODES) return;            // wave-uniform exit
  const int m    = lane & 15;
  const int half = lane >> 4;
  const int row  = row0 + m;
  const float ri = rin[row];
  const float ro = rout[row];
  const float* ar = AGG1 + (size_t)row * HID;

  v8f acc = {};
#pragma unroll
  for (int k = 0; k < HID; k += 4) {
    const int d0 = k + 2 * half, d1 = d0 + 1;
    v2f a; a[0] = (ar[d0] * ri + b0[d0]) * ro;
           a[1] = (ar[d1] * ri + b0[d1]) * ro;
    v2f b; b[0] = W1[d0 * EMBD + m];
           b[1] = W1[d1 * EMBD + m];
    acc = __builtin_amdgcn_wmma_f32_16x16x4_f32(false, a, false, b, (short)0, acc,
                                                false, false);
  }
  float* outp = H2 + (size_t)(row0 + 8 * half) * EMBD + m;
#pragma unroll
  for (int i = 0; i < 8; ++i) outp[(size_t)i * EMBD] = acc[i];
}

// ---------------- final: emb = agg2*rin + b1 ; out = emb @ Wl + bl ----------------
__global__ void k_final(const float* __restrict__ AGG2, const float* __restrict__ rin,
                        const float* __restrict__ b1, const float* __restrict__ Wl,
                        const float* __restrict__ bl, float* __restrict__ emb_out,
                        float* __restrict__ cls_out) {
  int i = blockIdx.x * blockDim.x + threadIdx.x;
  if (i >= N_NODES) return;
  const float ri = rin[i];
  const float* a = AGG2 + (size_t)i * EMBD;
  float e[EMBD];
  float o0 = bl[0], o1 = bl[1];
#pragma unroll
  for (int d = 0; d < EMBD; ++d) {
    e[d] = a[d] * ri + b1[d];
    o0 += e[d] * Wl[d * NCLS + 0];
    o1 += e[d] * Wl[d * NCLS + 1];
  }
  float* eo = emb_out + (size_t)i * EMBD;
#pragma unroll
  for (int d = 0; d < EMBD; ++d) eo[d] = e[d];
  cls_out[i * NCLS + 0] = o0;
  cls_out[i * NCLS + 1] = o1;
}

extern "C" void kernel_launch(void* const* d_in, const int* in_sizes, int n_in,
                              void* d_out, int out_size, void* d_ws, size_t ws_size,
                              hipStream_t stream) {
  (void)in_sizes; (void)n_in; (void)out_size; (void)ws_size;

  const int*   src = (const int*)d_in[0];
  const int*   dst = (const int*)d_in[1];
  const float* X   = (const float*)d_in[2];
  const float* W0  = (const float*)d_in[3];
  const float* b0  = (const float*)d_in[4];
  const float* W1  = (const float*)d_in[5];
  const float* b1  = (const float*)d_in[6];
  const float* Wl  = (const float*)d_in[7];
  const float* bl  = (const float*)d_in[8];

  float* ws     = (float*)d_ws;
  float* outdeg = ws + WS_OUTDEG;
  float* indeg  = ws + WS_INDEG;
  float* rout   = ws + WS_ROUT;
  float* rin    = ws + WS_RIN;
  float* H0     = ws + WS_H0;
  float* agg1   = ws + WS_AGG1;
  float* H2     = ws + WS_H2;
  float* agg2   = ws + WS_AGG2;

  float* emb_out = (float*)d_out;                       // 100000 x 16
  float* cls_out = (float*)d_out + (size_t)N_NODES * EMBD; // 100000 x 2

  const int B = 256;

  // 0) zero degree + aggregation scratch (zero whole scratch region; cheap)
  k_zero<<<(WS_TOTAL + B - 1) / B, B, 0, stream>>>(ws, WS_TOTAL);

  // 1) degrees
  k_degrees<<<(N_EDGES + B - 1) / B, B, 0, stream>>>(src, dst, outdeg, indeg);

  // 2) rsqrt normalizers
  k_rsqrt<<<(N_NODES + B - 1) / B, B, 0, stream>>>(outdeg, indeg, rout, rin);

  // 3) GEMM1 via f32 WMMA: 6250 waves, one 16x16 tile each
  {
    const int waves = N_NODES / 16;                // 6250, exact
    const int blocks = (waves * 32 + B - 1) / B;   // 782
    k_gemm1_wmma<<<blocks, B, 0, stream>>>(X, W0, rout, H0);
  }

  // 4) scatter layer 1
  k_scatter<<<(N_EDGES * 4 + B - 1) / B, B, 0, stream>>>(H0, src, dst, agg1);

  // 5) fused layer-1 epilogue + GEMM2 via f32 WMMA
  {
    const int waves = N_NODES / 16;
    const int blocks = (waves * 32 + B - 1) / B;
    k_gemm2_wmma<<<blocks, B, 0, stream>>>(agg1, W1, b0, rin, rout, H2);
  }

  // 6) scatter layer 2
  k_scatter<<<(N_EDGES * 4 + B - 1) / B, B, 0, stream>>>(H2, src, dst, agg2);

  // 7) final normalize + classifier
  k_final<<<(N_NODES + B - 1) / B, B, 0, stream>>>(agg2, rin, b1, Wl, bl,
                                                   emb_out, cls_out);
}